// MultiBoxLoss_40372692582723
// MI455X (gfx1250) — compile-verified
//
#include <hip/hip_runtime.h>
#include <hip/hip_bf16.h>
#include <stdint.h>

// Problem constants (from reference)
constexpr int kB = 8;
constexpr int kP = 100000;
constexpr int kC = 81;
constexpr int kN = 32;
constexpr int kNBlk    = (kP + 255) / 256;   // 391 blocks per batch row
constexpr int kNBlkAll = kB * kNBlk;         // 3128 total blocks

#define AS1 __attribute__((address_space(1)))
#define AS3 __attribute__((address_space(3)))

typedef int v4i __attribute__((ext_vector_type(4)));

#if __has_builtin(__builtin_amdgcn_global_load_async_to_lds_b128) && \
    __has_builtin(__builtin_amdgcn_global_load_async_to_lds_b32)
#define HAVE_ASYNC_LDS 1
#endif

// ---------------- workspace layout ----------------
struct WS {
  unsigned long long gtBest[kB * kN];                    // best prior per gt (packed ov|~p)
  unsigned long long blkBest[(size_t)kNBlkAll * kN];     // per-block partials of gtBest
  float bestOv[(size_t)kB * kP];                         // best_truth_ov
  int   bestIdx[(size_t)kB * kP];                        // best_truth_idx
  float lcall[(size_t)kB * kP];                          // masked loss_c_all
  int   posCnt[kB];
  int   posTot;
  int   hist1[kB * 2048];
  int   hist2[kB * 2048];
  int   hist3[kB * 1024];
  unsigned selPrefix[kB];
  int      selRem[kB];
  int      selK[kB];
  unsigned selT[kB];                                     // bits of k-th largest value
  int      selTie[kB];                                   // # ties at t included
  float partLossL[kNBlkAll];
  float partPosCE[kNBlkAll];
  float partNeg[kNBlkAll];
};

// ---------------- device helpers ----------------
__device__ __forceinline__ void wait_async0() {
#if __has_builtin(__builtin_amdgcn_s_wait_asynccnt)
  __builtin_amdgcn_s_wait_asynccnt(0);
#else
  asm volatile("s_wait_asynccnt 0" ::: "memory");
#endif
}

__device__ __forceinline__ float smooth_l1(float d) {
  d = fabsf(d);
  return d < 1.0f ? 0.5f * d * d : d - 0.5f;
}

__device__ __forceinline__ unsigned long long umax64(unsigned long long a,
                                                     unsigned long long b) {
  return a > b ? a : b;
}

// ---------------- kernels ----------------
__global__ void k_init(WS* __restrict__ w) {
  int i = blockIdx.x * blockDim.x + threadIdx.x;
  int nth = gridDim.x * blockDim.x;
  for (int j = i; j < kB * kN; j += nth) w->gtBest[j] = 0ull;
  for (int j = i; j < kB * 2048; j += nth) { w->hist1[j] = 0; w->hist2[j] = 0; }
  for (int j = i; j < kB * 1024; j += nth) w->hist3[j] = 0;
  if (i < kB) w->posCnt[i] = 0;
  if (i == 0) w->posTot = 0;
}

// IoU of every prior against all N gts: per-prior max/argmax (truth side)
// + per-gt max/argmax over priors via wave shuffles + block slots (prior side).
__global__ void k_iou(const float* __restrict__ priors,
                      const float* __restrict__ gtb,
                      const int* __restrict__ gtl,
                      WS* __restrict__ w) {
  const int blk = blockIdx.x;
  const int b = blk / kNBlk, pb = blk % kNBlk;
  const int tid = threadIdx.x;
  const int p = pb * 256 + tid;
  const bool act = p < kP;

  __shared__ __align__(16) float sB[kN * 4];
  __shared__ int sL[kN];
  __shared__ unsigned long long sBest[8][kN];

#ifdef HAVE_ASYNC_LDS
  if (tid < kN) {
    // gfx1250 async global->LDS copy (ASYNCcnt path)
    __builtin_amdgcn_global_load_async_to_lds_b128(
        (AS1 v4i*)(uintptr_t)(gtb + ((size_t)b * kN + tid) * 4),
        (AS3 v4i*)(unsigned)(uintptr_t)(&sB[tid * 4]), 0, 0);
    __builtin_amdgcn_global_load_async_to_lds_b32(
        (AS1 int*)(uintptr_t)(gtl + (size_t)b * kN + tid),
        (AS3 int*)(unsigned)(uintptr_t)(&sL[tid]), 0, 0);
  }
  if (tid < 32) wait_async0();   // wave0 issued; drain ASYNCcnt before barrier
  __syncthreads();
#else
  if (tid < kN) {
    sB[tid * 4 + 0] = gtb[((size_t)b * kN + tid) * 4 + 0];
    sB[tid * 4 + 1] = gtb[((size_t)b * kN + tid) * 4 + 1];
    sB[tid * 4 + 2] = gtb[((size_t)b * kN + tid) * 4 + 2];
    sB[tid * 4 + 3] = gtb[((size_t)b * kN + tid) * 4 + 3];
    sL[tid] = gtl[(size_t)b * kN + tid];
  }
  __syncthreads();
#endif

  float ax1 = 0.f, ay1 = 0.f, ax2 = 0.f, ay2 = 0.f, areaB = 0.f;
  if (act) {
    float px = priors[(size_t)p * 4 + 0], py = priors[(size_t)p * 4 + 1];
    float pw = priors[(size_t)p * 4 + 2], ph = priors[(size_t)p * 4 + 3];
    ax1 = px - pw * 0.5f; ay1 = py - ph * 0.5f;
    ax2 = px + pw * 0.5f; ay2 = py + ph * 0.5f;
    areaB = (ax2 - ax1) * (ay2 - ay1);
  }

  float bestV = -2.0f;
  int bestN = 0;
  for (int n = 0; n < kN; ++n) {
    float gx1 = sB[n * 4 + 0], gy1 = sB[n * 4 + 1];
    float gx2 = sB[n * 4 + 2], gy2 = sB[n * 4 + 3];
    bool valid = sL[n] > 0;
    float tlx = fmaxf(gx1, ax1), tly = fmaxf(gy1, ay1);
    float brx = fminf(gx2, ax2), bry = fminf(gy2, ay2);
    float iw = fmaxf(brx - tlx, 0.f), ih = fmaxf(bry - tly, 0.f);
    float inter = iw * ih;
    float areaA = (gx2 - gx1) * (gy2 - gy1);
    float ov = inter / (areaA + areaB - inter + 1e-9f);
    ov = valid ? ov : -1.0f;                 // invalid gt row -> -1 (matches ref)
    if (act && ov > bestV) { bestV = ov; bestN = n; }  // first-max-wins like argmax

    // prior-side: pack (ov_bits, ~p) so max => largest ov, ties -> smallest p
    unsigned long long pk = 0ull;
    if (act && valid)
      pk = ((unsigned long long)__float_as_uint(ov) << 32) |
           (unsigned long long)(0xFFFFFFFFu - (unsigned)p);
    for (int off = 16; off > 0; off >>= 1)
      pk = umax64(pk, (unsigned long long)__shfl_xor((unsigned long long)pk, off));
    if ((tid & 31) == 0) sBest[tid >> 5][n] = pk;
  }
  __syncthreads();

  if (tid < kN) {
    unsigned long long m = 0ull;
    for (int wv = 0; wv < 8; ++wv) m = umax64(m, sBest[wv][tid]);
    w->blkBest[(size_t)blk * kN + tid] = m;   // no atomics: deterministic
  }
  if (act) {
    w->bestOv[(size_t)b * kP + p] = bestV;
    w->bestIdx[(size_t)b * kP + p] = bestN;
  }
}

__global__ void k_gtreduce(WS* __restrict__ w) {
  int tid = threadIdx.x;            // 256 threads = (b, n)
  int b = tid / kN, n = tid % kN;
  unsigned long long m = 0ull;
  for (int j = 0; j < kNBlk; ++j)
    m = umax64(m, w->blkBest[((size_t)(b * kNBlk + j)) * kN + n]);
  w->gtBest[tid] = m;
  (void)b;
}

// Force best prior of each valid gt: ov=2.0, idx=n (sequential per batch => last-n wins)
__global__ void k_scatter(const int* __restrict__ gtl, WS* __restrict__ w) {
  int b = threadIdx.x;
  if (b >= kB) return;
  for (int n = 0; n < kN; ++n) {
    if (gtl[b * kN + n] > 0) {
      unsigned long long pk = w->gtBest[b * kN + n];
      unsigned p = 0xFFFFFFFFu - (unsigned)(pk & 0xFFFFFFFFull);
      if (p < (unsigned)kP) {
        w->bestOv[(size_t)b * kP + p] = 2.0f;
        w->bestIdx[(size_t)b * kP + p] = n;
      }
    }
  }
}

// Per (b,p): conf target, logsumexp over C, loss_c_all, smooth-L1 for positives.
__global__ void k_conf(const float* __restrict__ loc,
                       const float* __restrict__ conf,
                       const float* __restrict__ priors,
                       const float* __restrict__ gtb,
                       const int* __restrict__ gtl,
                       WS* __restrict__ w) {
  const int blk = blockIdx.x;
  const int b = blk / kNBlk, pb = blk % kNBlk;
  const int tid = threadIdx.x;
  const int p = pb * 256 + tid;
  const bool act = p < kP;

  float myLossL = 0.f, myPosCE = 0.f;
  int myPos = 0;

  if (act) {
    size_t bp = (size_t)b * kP + p;
    float ov = w->bestOv[bp];
    int n = w->bestIdx[bp];
    int cf = gtl[b * kN + n];
    if (ov < 0.5f) cf = -1;      // neutral band
    if (ov < 0.4f) cf = 0;       // background
    int tgt = cf > 0 ? cf : 0;

    const float* row = conf + bp * (size_t)kC;
    float m = -3.4e38f;
    #pragma unroll 3
    for (int c = 0; c < kC; ++c) m = fmaxf(m, row[c]);
    float s = 0.f;
    #pragma unroll 3
    for (int c = 0; c < kC; ++c) s += __expf(row[c] - m);
    float lse = m + __logf(s);
    float ce = lse - row[tgt];

    bool pos = cf > 0;
    w->lcall[bp] = (pos || cf < 0) ? 0.f : ce;   // masked mining score (>=0)

    if (pos) {
      myPos = 1;
      myPosCE = ce;
      float gx1 = gtb[((size_t)b * kN + n) * 4 + 0];
      float gy1 = gtb[((size_t)b * kN + n) * 4 + 1];
      float gx2 = gtb[((size_t)b * kN + n) * 4 + 2];
      float gy2 = gtb[((size_t)b * kN + n) * 4 + 3];
      float pcx = priors[(size_t)p * 4 + 0], pcy = priors[(size_t)p * 4 + 1];
      float pw  = priors[(size_t)p * 4 + 2], ph  = priors[(size_t)p * 4 + 3];
      float t0 = ((gx1 + gx2) * 0.5f - pcx) / (0.1f * pw);
      float t1 = ((gy1 + gy2) * 0.5f - pcy) / (0.1f * ph);
      float t2 = __logf(fmaxf((gx2 - gx1) / pw, 1e-6f)) / 0.2f;
      float t3 = __logf(fmaxf((gy2 - gy1) / ph, 1e-6f)) / 0.2f;
      myLossL = smooth_l1(loc[bp * 4 + 0] - t0) + smooth_l1(loc[bp * 4 + 1] - t1) +
                smooth_l1(loc[bp * 4 + 2] - t2) + smooth_l1(loc[bp * 4 + 3] - t3);
    }
  }

  __shared__ float red[256];
  __shared__ int redi[256];
  red[tid] = myLossL; redi[tid] = myPos;
  __syncthreads();
  for (int s = 128; s > 0; s >>= 1) {
    if (tid < s) { red[tid] += red[tid + s]; redi[tid] += redi[tid + s]; }
    __syncthreads();
  }
  if (tid == 0) {
    w->partLossL[blk] = red[0];
    atomicAdd(&w->posCnt[b], redi[0]);   // integer atomics: deterministic
    atomicAdd(&w->posTot, redi[0]);
  }
  __syncthreads();
  red[tid] = myPosCE;
  __syncthreads();
  for (int s = 128; s > 0; s >>= 1) {
    if (tid < s) red[tid] += red[tid + s];
    __syncthreads();
  }
  if (tid == 0) w->partPosCE[blk] = red[0];
}

// ----- 3-round radix select of k-th largest (values >= 0 => bits monotone) -----
__global__ void k_hist1(WS* __restrict__ w) {
  int blk = blockIdx.x, b = blk / kNBlk, pb = blk % kNBlk;
  int p = pb * 256 + threadIdx.x;
  if (p >= kP) return;
  unsigned bits = __float_as_uint(w->lcall[(size_t)b * kP + p]);
  atomicAdd(&w->hist1[b * 2048 + (bits >> 21)], 1);
}

__global__ void k_sel1(WS* __restrict__ w) {
  int b = threadIdx.x;
  if (b >= kB) return;
  int k = 3 * w->posCnt[b];
  if (k > kP - 1) k = kP - 1;
  w->selK[b] = k;
  if (k <= 0) return;
  int rem = k, bin = 2047;
  for (; bin >= 0; --bin) {
    int c = w->hist1[b * 2048 + bin];
    if (rem <= c) break;
    rem -= c;
  }
  if (bin < 0) bin = 0;
  w->selPrefix[b] = (unsigned)bin << 21;
  w->selRem[b] = rem;
}

__global__ void k_hist2(WS* __restrict__ w) {
  int blk = blockIdx.x, b = blk / kNBlk, pb = blk % kNBlk;
  int p = pb * 256 + threadIdx.x;
  if (p >= kP || w->selK[b] <= 0) return;
  unsigned bits = __float_as_uint(w->lcall[(size_t)b * kP + p]);
  if ((bits >> 21) == (w->selPrefix[b] >> 21))
    atomicAdd(&w->hist2[b * 2048 + ((bits >> 10) & 0x7FFu)], 1);
}

__global__ void k_sel2(WS* __restrict__ w) {
  int b = threadIdx.x;
  if (b >= kB || w->selK[b] <= 0) return;
  int rem = w->selRem[b], bin = 2047;
  for (; bin >= 0; --bin) {
    int c = w->hist2[b * 2048 + bin];
    if (rem <= c) break;
    rem -= c;
  }
  if (bin < 0) bin = 0;
  w->selPrefix[b] |= (unsigned)bin << 10;
  w->selRem[b] = rem;
}

__global__ void k_hist3(WS* __restrict__ w) {
  int blk = blockIdx.x, b = blk / kNBlk, pb = blk % kNBlk;
  int p = pb * 256 + threadIdx.x;
  if (p >= kP || w->selK[b] <= 0) return;
  unsigned bits = __float_as_uint(w->lcall[(size_t)b * kP + p]);
  if ((bits >> 10) == (w->selPrefix[b] >> 10))
    atomicAdd(&w->hist3[b * 1024 + (bits & 0x3FFu)], 1);
}

__global__ void k_sel3(WS* __restrict__ w) {
  int b = threadIdx.x;
  if (b >= kB || w->selK[b] <= 0) return;
  int rem = w->selRem[b], bin = 1023;
  for (; bin >= 0; --bin) {
    int c = w->hist3[b * 1024 + bin];
    if (rem <= c) break;
    rem -= c;
  }
  if (bin < 0) bin = 0;
  w->selT[b] = w->selPrefix[b] | (unsigned)bin;  // exact k-th largest value bits
  w->selTie[b] = rem;                            // # ties at t to include
}

// Sum of mining scores strictly above threshold t (per block partial).
__global__ void k_neg(WS* __restrict__ w) {
  int blk = blockIdx.x, b = blk / kNBlk, pb = blk % kNBlk;
  int tid = threadIdx.x, p = pb * 256 + tid;
  float v = 0.f;
  if (p < kP && w->selK[b] > 0) {
    float x = w->lcall[(size_t)b * kP + p];
    if (x > __uint_as_float(w->selT[b])) v = x;
  }
  __shared__ float red[256];
  red[tid] = v;
  __syncthreads();
  for (int s = 128; s > 0; s >>= 1) {
    if (tid < s) red[tid] += red[tid + s];
    __syncthreads();
  }
  if (tid == 0) w->partNeg[blk] = red[0];
}

__global__ void k_final(WS* __restrict__ w, float* __restrict__ out) {
  __shared__ float red[256];
  __shared__ float tot[3];
  int tid = threadIdx.x;
  float a = 0.f, bb = 0.f, cc = 0.f;
  for (int j = tid; j < kNBlkAll; j += 256) {
    a += w->partLossL[j];
    bb += w->partPosCE[j];
    cc += w->partNeg[j];
  }
  float vals[3] = {a, bb, cc};
  for (int v = 0; v < 3; ++v) {
    red[tid] = vals[v];
    __syncthreads();
    for (int s = 128; s > 0; s >>= 1) {
      if (tid < s) red[tid] += red[tid + s];
      __syncthreads();
    }
    if (tid == 0) tot[v] = red[0];
    __syncthreads();
  }
  if (tid == 0) {
    float tie = 0.f;
    for (int b = 0; b < kB; ++b)
      if (w->selK[b] > 0)
        tie += (float)w->selTie[b] * __uint_as_float(w->selT[b]);
    float denom = (float)(w->posTot > 1 ? w->posTot : 1);
    out[0] = tot[0] / denom;
    out[1] = (tot[1] + tot[2] + tie) / denom;
  }
}

// ---------------- host entry ----------------
extern "C" void kernel_launch(void* const* d_in, const int* in_sizes, int n_in,
                              void* d_out, int out_size, void* d_ws, size_t ws_size,
                              hipStream_t stream) {
  const float* loc    = (const float*)d_in[0];   // [B,P,4]
  const float* conf   = (const float*)d_in[1];   // [B,P,C]
  const float* priors = (const float*)d_in[2];   // [P,4] cxcywh
  const float* gtb    = (const float*)d_in[3];   // [B,N,4] xyxy
  const int*   gtl    = (const int*)d_in[4];     // [B,N]
  WS* w = (WS*)d_ws;
  float* out = (float*)d_out;

  k_init<<<64, 256, 0, stream>>>(w);
  k_iou<<<kNBlkAll, 256, 0, stream>>>(priors, gtb, gtl, w);
  k_gtreduce<<<1, 256, 0, stream>>>(w);
  k_scatter<<<1, 32, 0, stream>>>(gtl, w);
  k_conf<<<kNBlkAll, 256, 0, stream>>>(loc, conf, priors, gtb, gtl, w);
  k_hist1<<<kNBlkAll, 256, 0, stream>>>(w);
  k_sel1<<<1, 32, 0, stream>>>(w);
  k_hist2<<<kNBlkAll, 256, 0, stream>>>(w);
  k_sel2<<<1, 32, 0, stream>>>(w);
  k_hist3<<<kNBlkAll, 256, 0, stream>>>(w);
  k_sel3<<<1, 32, 0, stream>>>(w);
  k_neg<<<kNBlkAll, 256, 0, stream>>>(w);
  k_final<<<1, 256, 0, stream>>>(w, out);

  (void)in_sizes; (void)n_in; (void)out_size; (void)ws_size;
}